// EmbeddingLayer_7000796693061
// MI455X (gfx1250) — compile-verified
//
#include <hip/hip_runtime.h>

// Embedding gather: out[token, :] = table[idx[token], :]
// E = 128 fp32 = 512 bytes per row = 32 lanes x 16B  -> one wave32 per token row.
// Pure bandwidth problem (zero useful FLOPs; WMMA inapplicable): use B128
// loads, non-temporal B128 stores (keep the 16.4 MB table resident in the
// 192 MB L2 across graph replays), and a speculative global_prefetch_b8 on
// the index stream.

#define EMBED 128
#define LANES_PER_ROW 32  // EMBED * 4B / 16B = 32 == wave32

// Native clang vector type: valid for __builtin_nontemporal_store and
// lowers to single b128 memory ops (HIP's float4 class type does not).
typedef __attribute__((ext_vector_type(4))) float v4f;

__global__ __launch_bounds__(256) void embedding_gather_kernel(
    const int* __restrict__ idx,
    const v4f* __restrict__ table4,   // table viewed as [VOCAB][32] v4f
    v4f* __restrict__ out4,           // out  viewed as [NTOK][32] v4f
    int n_tokens) {
  const int tid   = blockIdx.x * blockDim.x + threadIdx.x;
  const int token = tid >> 5;   // one wave32 per token row
  const int lane  = tid & 31;   // lane covers 16 bytes of the 512-byte row

  if (token >= n_tokens) return;

  // Speculative prefetch of the index cacheline ~1 KB ahead.
  // gfx1250: lowers to global_prefetch_b8; OOB translation failures are
  // silently dropped for speculative prefetches.
  __builtin_prefetch(idx + token + 256, /*rw=*/0, /*locality=*/1);

  // Wave-uniform index load: all 32 lanes request the same address ->
  // a single cacheline transaction per wave.
  const int row = idx[token];

  // Coalesced 512-byte row move: global_load_b128 ...
  const v4f v = table4[(size_t)row * LANES_PER_ROW + lane];

  // ... + non-temporal global_store_b128 (TH=NT): output is write-once
  // streaming data; do not let it evict the L2-resident embedding table.
  __builtin_nontemporal_store(v, out4 + (size_t)token * LANES_PER_ROW + lane);
}

extern "C" void kernel_launch(void* const* d_in, const int* in_sizes, int n_in,
                              void* d_out, int out_size, void* d_ws, size_t ws_size,
                              hipStream_t stream) {
  (void)n_in; (void)d_ws; (void)ws_size; (void)out_size;

  const int*   idx   = (const int*)d_in[0];      // x: [B, S] int32
  const float* table = (const float*)d_in[1];    // embedding_table: [VOCAB, EMBED] f32
  float*       out   = (float*)d_out;            // [B, S, EMBED] f32

  const int n_tokens  = in_sizes[0];                  // B * S = 8192
  const int n_threads = n_tokens * LANES_PER_ROW;     // one lane per 16B chunk
  const int block     = 256;                          // 8 wave32s per block
  const int grid      = (n_threads + block - 1) / block;

  embedding_gather_kernel<<<grid, block, 0, stream>>>(
      idx, (const v4f*)table, (v4f*)out, n_tokens);
}